// HierarchicalSoftMatcher_53188874994218
// MI455X (gfx1250) — compile-verified
//
#include <hip/hip_runtime.h>
#include <cmath>

// Problem constants (match reference)
#define NPTS   2048
#define CDIM   256
#define NBATCH 2
#define KC     128
#define KF     16
#define SLAB   512
#define NEGV   (-1.0e9f)
#define LOCRAD 0.26179938779914946f   // radians(15)

typedef float v2f __attribute__((ext_vector_type(2)));
typedef float v8f __attribute__((ext_vector_type(8)));

// Monotonic float -> uint mapping (larger float => larger uint)
__device__ __forceinline__ unsigned f2ord(float f) {
  unsigned u = __float_as_uint(f);
  return (u & 0x80000000u) ? ~u : (u | 0x80000000u);
}

// ---------------------------------------------------------------------------
// Kernel 1: q = feat_a @ Wq^T + bq ; k = feat_b @ Wk^T + bk  (fp32 WMMA)
// One wave computes a 16x64 strip (4 col tiles, 4 independent accumulators):
// A fragment reused 4x, 4 independent WMMA chains hide XDL latency.
// 2 matrices * 256 row tiles * 4 col groups = 2048 waves -> 256 blocks.
// ---------------------------------------------------------------------------
__global__ __launch_bounds__(256)
void proj_kernel(const float* __restrict__ featA, const float* __restrict__ featB,
                 const float* __restrict__ Wq, const float* __restrict__ bq,
                 const float* __restrict__ Wk, const float* __restrict__ bk,
                 float* __restrict__ q, float* __restrict__ k)
{
  const int wid  = threadIdx.x >> 5;
  const int lane = threadIdx.x & 31;
  const int tile = blockIdx.x * 8 + wid;          // 0..2047
  const int which = tile >> 10;                   // 0 => q, 1 => k
  const int t  = tile & 1023;
  const int mt = t >> 2;                          // row tile 0..255
  const int ng = t & 3;                           // col group 0..3 (4 tiles each)

  const float* X    = which ? featB : featA;
  const float* W    = which ? Wk    : Wq;
  const float* bias = which ? bk    : bq;
  float*       O    = which ? k     : q;

  const int l15  = lane & 15;
  const int half = lane >> 4;                     // 0 or 1
  const int mrow = (mt << 4) + l15;               // A: M = lane index
  const int nc0  = ((ng << 2) << 4) + l15;        // first col tile's N index

  // A frag: lane<16 holds K={k0,k0+1}, lane>=16 holds K={k0+2,k0+3}
  const float* aptr = X + (size_t)mrow * CDIM + half * 2;
  // B[kk][n] = W[n][k0+kk]  (half-split mirrors A)
  const float* bptr = W + (size_t)nc0 * CDIM + half * 2;

  v8f acc0 = {0.f,0.f,0.f,0.f,0.f,0.f,0.f,0.f};
  v8f acc1 = acc0, acc2 = acc0, acc3 = acc0;
  for (int k0 = 0; k0 < CDIM; k0 += 4) {
    v2f a  = *(const v2f*)(aptr + k0);
    v2f b0 = *(const v2f*)(bptr + k0);
    v2f b1 = *(const v2f*)(bptr + 16 * CDIM + k0);
    v2f b2 = *(const v2f*)(bptr + 32 * CDIM + k0);
    v2f b3 = *(const v2f*)(bptr + 48 * CDIM + k0);
    acc0 = __builtin_amdgcn_wmma_f32_16x16x4_f32(false, a, false, b0, (short)0, acc0, false, false);
    acc1 = __builtin_amdgcn_wmma_f32_16x16x4_f32(false, a, false, b1, (short)0, acc1, false, false);
    acc2 = __builtin_amdgcn_wmma_f32_16x16x4_f32(false, a, false, b2, (short)0, acc2, false, false);
    acc3 = __builtin_amdgcn_wmma_f32_16x16x4_f32(false, a, false, b3, (short)0, acc3, false, false);
  }
  // D layout: VGPR v, lane<16 -> M=v ; lane>=16 -> M=v+8
  const int obase = ((mt << 4) + (half << 3)) * CDIM + nc0;
#pragma unroll
  for (int g = 0; g < 4; ++g) {
    const v8f acc = (g == 0) ? acc0 : (g == 1) ? acc1 : (g == 2) ? acc2 : acc3;
    const float bv = bias[nc0 + g * 16];
#pragma unroll
    for (int v = 0; v < 8; ++v)
      O[obase + g * 16 + v * CDIM] = acc[v] + bv;
  }
}

// ---------------------------------------------------------------------------
// Kernel 2: sim slab — sim[gr, j] = q[gr,:] . k[b, j, :]  (fp32 WMMA)
// One wave per 16x64 strip of a 512 x 2048 slab (slabs never cross batches).
// 32 row tiles * 32 col groups = 1024 waves -> 128 blocks.
// ---------------------------------------------------------------------------
__global__ __launch_bounds__(256)
void sim_kernel(const float* __restrict__ q, const float* __restrict__ k,
                float* __restrict__ simSlab, int slabBase)
{
  const int wid  = threadIdx.x >> 5;
  const int lane = threadIdx.x & 31;
  const int tile = blockIdx.x * 8 + wid;          // 0..1023
  const int mt = tile >> 5;                       // 0..31 (row tile in slab)
  const int ng = tile & 31;                       // 0..31 (col group of 4 tiles)

  const int gr0 = slabBase + (mt << 4);
  const int b   = gr0 / NPTS;
  const int l15  = lane & 15;
  const int half = lane >> 4;
  const int jc0  = ((ng << 2) << 4) + l15;        // first key index of strip

  const float* aptr = q + (size_t)(gr0 + l15) * CDIM + half * 2;
  const float* bptr = k + (size_t)(b * NPTS + jc0) * CDIM + half * 2;

  v8f acc0 = {0.f,0.f,0.f,0.f,0.f,0.f,0.f,0.f};
  v8f acc1 = acc0, acc2 = acc0, acc3 = acc0;
  for (int k0 = 0; k0 < CDIM; k0 += 4) {
    v2f a  = *(const v2f*)(aptr + k0);
    v2f b0 = *(const v2f*)(bptr + k0);
    v2f b1 = *(const v2f*)(bptr + 16 * CDIM + k0);
    v2f b2 = *(const v2f*)(bptr + 32 * CDIM + k0);
    v2f b3 = *(const v2f*)(bptr + 48 * CDIM + k0);
    acc0 = __builtin_amdgcn_wmma_f32_16x16x4_f32(false, a, false, b0, (short)0, acc0, false, false);
    acc1 = __builtin_amdgcn_wmma_f32_16x16x4_f32(false, a, false, b1, (short)0, acc1, false, false);
    acc2 = __builtin_amdgcn_wmma_f32_16x16x4_f32(false, a, false, b2, (short)0, acc2, false, false);
    acc3 = __builtin_amdgcn_wmma_f32_16x16x4_f32(false, a, false, b3, (short)0, acc3, false, false);
  }
  const int orow0 = (mt << 4) + (half << 3);
#pragma unroll
  for (int g = 0; g < 4; ++g) {
    const v8f acc = (g == 0) ? acc0 : (g == 1) ? acc1 : (g == 2) ? acc2 : acc3;
#pragma unroll
    for (int v = 0; v < 8; ++v)
      simSlab[(size_t)(orow0 + v) * NPTS + jc0 + g * 16] = acc[v];
  }
}

// ---------------------------------------------------------------------------
// Kernel 3: per-row top-128 radix-select -> angular mask (+fallback) ->
//           top-16 -> masked softmax -> scatter into dense P row.
// One block (256 threads) per query row.
// ---------------------------------------------------------------------------
__global__ __launch_bounds__(256)
void topk_kernel(const float* __restrict__ simSlab,
                 const float* __restrict__ dirsA, const float* __restrict__ dirsB,
                 float* __restrict__ P, int slabBase)
{
  __shared__ float    sSim[NPTS];        // 8 KB: full sim row
  __shared__ unsigned sHist[256];
  __shared__ int      sScan[256];
  __shared__ int      sCand[KC];
  __shared__ float    sLogit[KC];
  __shared__ float    sWork[KC];
  __shared__ int      sMask[KC];
  __shared__ float    sRed[KC];
  __shared__ int      sRedIdx[KC];
  __shared__ int      sSel[KF];
  __shared__ float    sProb[KF];
  __shared__ int      sCounters[2];      // [0]: >T slot counter, [1]: any(mask)
  __shared__ unsigned sThresh[2];        // prefix, remaining

  const int tid = threadIdx.x;
  const int lr  = blockIdx.x;            // local row in slab
  const int gr  = slabBase + lr;         // global row (b*NPTS + i)
  const int b   = gr / NPTS;

  if (tid == 0) { sCounters[0] = 0; sCounters[1] = 0; }
  for (int j = tid; j < NPTS; j += 256)
    sSim[j] = simSlab[(size_t)lr * NPTS + j];
  __syncthreads();

  // ---- radix-select the 128th largest key (MSB-first, 8 bits/pass) ----
  unsigned prefix = 0, remaining = KC;
  for (int shift = 24; shift >= 0; shift -= 8) {
    sHist[tid] = 0;
    __syncthreads();
    const unsigned hmask = (shift == 24) ? 0u : (0xFFFFFFFFu << (shift + 8));
#pragma unroll
    for (int e = 0; e < 8; ++e) {
      const unsigned u = f2ord(sSim[tid * 8 + e]);
      if ((u & hmask) == prefix)
        atomicAdd(&sHist[(u >> shift) & 255u], 1u);
    }
    __syncthreads();
    if (tid == 0) {
      unsigned cum = 0; int d = 255;
      for (; d > 0; --d) {
        const unsigned c = sHist[d];
        if (cum + c >= remaining) break;
        cum += c;
      }
      sThresh[0] = prefix | ((unsigned)d << shift);
      sThresh[1] = remaining - cum;
    }
    __syncthreads();
    prefix    = sThresh[0];
    remaining = sThresh[1];
  }
  const unsigned T = prefix;            // exact 128th-largest key
  const int nG = KC - (int)remaining;   // count of keys strictly greater

  // ---- collect candidates: all >T (order-free), ties ==T by ascending index
  int cntEq = 0;
#pragma unroll
  for (int e = 0; e < 8; ++e) {
    const int j = tid * 8 + e;
    const unsigned u = f2ord(sSim[j]);
    if (u > T)       { const int s = atomicAdd(&sCounters[0], 1); sCand[s] = j; }
    else if (u == T) { cntEq++; }
  }
  sScan[tid] = cntEq;
  __syncthreads();
  for (int off = 1; off < 256; off <<= 1) {   // inclusive Hillis-Steele scan
    const int v   = sScan[tid];
    const int add = (tid >= off) ? sScan[tid - off] : 0;
    __syncthreads();
    sScan[tid] = v + add;
    __syncthreads();
  }
  const int base = sScan[tid] - cntEq;        // exclusive prefix
  int rank = 0;
#pragma unroll
  for (int e = 0; e < 8; ++e) {
    const int j = tid * 8 + e;
    if (f2ord(sSim[j]) == T) {
      const int pos = base + rank;
      if (pos < (int)remaining) sCand[nG + pos] = j;
      rank++;
    }
  }
  __syncthreads();

  // ---- angular locality mask + logits ----
  const float dax = dirsA[gr * 3 + 0];
  const float day = dirsA[gr * 3 + 1];
  const float daz = dirsA[gr * 3 + 2];
  if (tid < KC) {
    const int j = sCand[tid];
    const float* db = dirsB + (size_t)(b * NPTS + j) * 3;
    float c = dax * db[0] + day * db[1] + daz * db[2];
    c = fminf(1.0f, fmaxf(-1.0f, c));
    const int m = (acosf(c) <= LOCRAD) ? 1 : 0;
    sMask[tid] = m;
    if (m) sCounters[1] = 1;                  // benign idempotent race
    sLogit[tid] = sSim[j] / 0.07f;
  }
  __syncthreads();
  const int anyMask = sCounters[1];
  if (tid < KC) {
    const int m = anyMask ? sMask[tid] : 1;   // row fallback: all valid
    sMask[tid] = m;
    if (!m) sLogit[tid] = NEGV;
    sWork[tid] = m ? sLogit[tid] : NEGV;
  }
  __syncthreads();

  // ---- fine stage: top-16 by iterated argmax over 128 entries ----
  for (int it = 0; it < KF; ++it) {
    if (tid < KC) { sRed[tid] = sWork[tid]; sRedIdx[tid] = tid; }
    __syncthreads();
    for (int off = KC / 2; off >= 1; off >>= 1) {
      if (tid < off) {
        const float a = sRed[tid], c = sRed[tid + off];
        const int  ia = sRedIdx[tid], ic = sRedIdx[tid + off];
        if (c > a || (c == a && ic < ia)) { sRed[tid] = c; sRedIdx[tid] = ic; }
      }
      __syncthreads();
    }
    if (tid == 0) {
      const int w = sRedIdx[0];
      sSel[it]  = w;
      sWork[w]  = -3.0e38f;                   // exclude from later rounds
    }
    __syncthreads();
  }

  // ---- masked softmax + row normalization (reference-exact) ----
  if (tid == 0) {
    float vals[KF]; int eff[KF]; int any = 0;
#pragma unroll
    for (int t = 0; t < KF; ++t) {
      const int w = sSel[t];
      vals[t] = sLogit[w];
      eff[t]  = sMask[w];
      any |= eff[t];
    }
    if (!any)
      for (int t = 0; t < KF; ++t) eff[t] = 1;
    float mx = -3.0e38f;
    float l[KF];
#pragma unroll
    for (int t = 0; t < KF; ++t) { l[t] = eff[t] ? vals[t] : NEGV; mx = fmaxf(mx, l[t]); }
    float s = 0.0f; float ex[KF];
#pragma unroll
    for (int t = 0; t < KF; ++t) { ex[t] = expf(l[t] - mx); s += ex[t]; }
    float rowsum = 0.0f;
#pragma unroll
    for (int t = 0; t < KF; ++t) {
      const float p = (ex[t] / s) * (eff[t] ? 1.0f : 0.0f);
      sProb[t] = p; rowsum += p;
    }
    const float inv = 1.0f / (rowsum + 1e-8f);
#pragma unroll
    for (int t = 0; t < KF; ++t) sProb[t] *= inv;
  }
  __syncthreads();

  // ---- write dense P row: zeros + 16 scattered probabilities ----
  float* prow = P + (size_t)gr * NPTS;
  for (int j = tid; j < NPTS; j += 256) prow[j] = 0.0f;
  __syncthreads();
  if (tid < KF) prow[sCand[sSel[tid]]] = sProb[tid];
}

// ---------------------------------------------------------------------------
extern "C" void kernel_launch(void* const* d_in, const int* in_sizes, int n_in,
                              void* d_out, int out_size, void* d_ws, size_t ws_size,
                              hipStream_t stream)
{
  (void)in_sizes; (void)n_in; (void)out_size; (void)ws_size;
  const float* featA = (const float*)d_in[0];
  const float* featB = (const float*)d_in[1];
  const float* dirsA = (const float*)d_in[2];
  const float* dirsB = (const float*)d_in[3];
  const float* Wq    = (const float*)d_in[4];
  const float* bq    = (const float*)d_in[5];
  const float* Wk    = (const float*)d_in[6];
  const float* bk    = (const float*)d_in[7];
  float* P = (float*)d_out;

  // Workspace layout: q (4 MB) | k (4 MB) | sim slab (4 MB) — all L2-resident.
  float* q       = (float*)d_ws;
  float* k       = q + (size_t)NBATCH * NPTS * CDIM;
  float* simSlab = k + (size_t)NBATCH * NPTS * CDIM;

  // 1) projections: 2048 wave-strips / 8 waves per block
  proj_kernel<<<256, 256, 0, stream>>>(featA, featB, Wq, bq, Wk, bk, q, k);

  // 2+3) per-slab: similarity GEMM then per-row selection/softmax/scatter
  const int nSlabs = (NBATCH * NPTS) / SLAB;   // 8
  for (int s = 0; s < nSlabs; ++s) {
    sim_kernel <<<128, 256, 0, stream>>>(q, k, simSlab, s * SLAB);
    topk_kernel<<<SLAB, 256, 0, stream>>>(simSlab, dirsA, dirsB, P, s * SLAB);
  }
}